// RWKV7TimeMixing_14310831030385
// MI455X (gfx1250) — compile-verified
//
#include <hip/hip_runtime.h>
#include <hip/hip_bf16.h>

typedef __attribute__((ext_vector_type(2))) float v2f;
typedef __attribute__((ext_vector_type(8))) float v8f;
typedef __attribute__((ext_vector_type(4))) unsigned int u32x4;
typedef __attribute__((ext_vector_type(8))) int i32x8;
typedef __attribute__((ext_vector_type(4))) int i32x4;

#define Cdim 256
#define Tdim 512
#define Bdim 8
#define BTC (Bdim * Tdim * Cdim)

__device__ __forceinline__ v8f wmma_f32(v2f a, v2f b, v8f c) {
  // D = A(16x4 f32) x B(4x16 f32) + C(16x16 f32)
  return __builtin_amdgcn_wmma_f32_16x16x4_f32(
      /*neg_a=*/false, a, /*neg_b=*/false, b,
      /*c_mod=*/(short)0, c, /*reuse_a=*/false, /*reuse_b=*/false);
}

// ---------------------------------------------------------------------------
// Tensor Data Mover: 2D tile load Global -> LDS (f32 elements).
// Descriptor per CDNA5 ISA ch.8: group0 = {flags, lds_addr, global_addr, type},
// group1 = {data_size, tensor dims, tile dims, dim0 stride}. 1 wave issues.
// This toolchain's builtin is the 6-arg clang-23 form:
//   (u32x4 g0, i32x8 g1, i32x4 g2, i32x4 g3, i32x8 extra, i32 cpol)
// ---------------------------------------------------------------------------
__device__ __forceinline__ void tdm_load_2d_f32(unsigned int lds_off,
                                                const void* gptr,
                                                unsigned int tile_d0,
                                                unsigned int tile_d1,
                                                unsigned int tensor_d0,
                                                unsigned int tensor_d1,
                                                unsigned int stride0) {
  unsigned long long ga = (unsigned long long)(uintptr_t)gptr;
  u32x4 g0;
  g0[0] = 1u;                                                  // count=1, user D#
  g0[1] = lds_off;                                             // LDS byte addr
  g0[2] = (unsigned int)ga;                                    // global addr lo
  g0[3] = ((unsigned int)(ga >> 32) & 0x01FFFFFFu) | (2u << 30);  // hi + type=2
  i32x8 g1;
  g1[0] = (int)(2u << 16);                                     // data_size=4B
  g1[1] = (int)((tensor_d0 & 0xFFFFu) << 16);                  // tensor_dim0 lo
  g1[2] = (int)((tensor_d0 >> 16) | ((tensor_d1 & 0xFFFFu) << 16));
  g1[3] = (int)((tensor_d1 >> 16) | (tile_d0 << 16));          // tile_dim0
  g1[4] = (int)(tile_d1 & 0xFFFFu);                            // tile_dim1
  g1[5] = (int)stride0;                                        // dim0 stride lo
  g1[6] = 0;
  g1[7] = 0;
  i32x4 z4 = {0, 0, 0, 0};
  i32x8 z8 = {0, 0, 0, 0, 0, 0, 0, 0};
  __builtin_amdgcn_tensor_load_to_lds(g0, g1, z4, z4, z8, 0);
}

__device__ __forceinline__ unsigned int lds_offset_of(const void* p) {
  // LDS offset is taken as the low 32 bits of the flat address
  return (unsigned int)(uintptr_t)p;
}

// ---------------------------------------------------------------------------
// LayerNorm: one block per row of x [B*T, C], C == blockDim == 256
// ---------------------------------------------------------------------------
__global__ __launch_bounds__(256) void ln_kernel(const float* __restrict__ x,
                                                 const float* __restrict__ gamma,
                                                 const float* __restrict__ beta,
                                                 float* __restrict__ xn) {
  __shared__ float red[256];
  const int row = blockIdx.x;
  const int tid = threadIdx.x;
  float v = x[row * Cdim + tid];
  red[tid] = v;
  __syncthreads();
  for (int s = 128; s > 0; s >>= 1) {
    if (tid < s) red[tid] += red[tid + s];
    __syncthreads();
  }
  float mu = red[0] * (1.0f / Cdim);
  __syncthreads();
  float d = v - mu;
  red[tid] = d * d;
  __syncthreads();
  for (int s = 128; s > 0; s >>= 1) {
    if (tid < s) red[tid] += red[tid + s];
    __syncthreads();
  }
  float var = red[0] * (1.0f / Cdim);
  xn[row * Cdim + tid] = d * rsqrtf(var + 1e-5f) * gamma[tid] + beta[tid];
}

// ---------------------------------------------------------------------------
// out[m,n] = sum_k A[m,k] * W[n,k]   (A: [M,256] row-major, W: [256,256])
// Block: 256 threads (8 waves). Block tile: 128(M) x 32(N).
// W n-panel (32 rows x 256) staged in LDS by the TDM; each A fragment feeds
// two WMMAs (two B columns) to double math per byte of A traffic.
// ---------------------------------------------------------------------------
__global__ __launch_bounds__(256) void gemm_xwt(const float* __restrict__ A,
                                                const float* __restrict__ W,
                                                float* __restrict__ out) {
  __shared__ float Wl[32 * Cdim];
  const int n0 = blockIdx.y * 32;
  const int m0 = blockIdx.x * 128;
  const int tid = threadIdx.x;

  if (tid < 32) {
    tdm_load_2d_f32(lds_offset_of(Wl), W + (size_t)n0 * Cdim,
                    /*tile*/ 256, 32, /*tensor*/ 256, 256, /*stride0*/ 256);
    __builtin_amdgcn_s_wait_tensorcnt(0);
  }
  __syncthreads();

  const int wave = tid >> 5, lane = tid & 31;
  const int nl = lane & 15;            // B column / A row (per-role)
  const int kh = (lane >> 4) * 2;      // K sub-offset for this half-wave
  const int mrow = m0 + wave * 16 + nl;

  v8f acc0 = {}, acc1 = {};
#pragma unroll 4
  for (int k0 = 0; k0 < 256; k0 += 4) {
    v2f a = *(const v2f*)(A + mrow * Cdim + k0 + kh);
    v2f b0 = *(const v2f*)(Wl + nl * Cdim + k0 + kh);
    v2f b1 = *(const v2f*)(Wl + (nl + 16) * Cdim + k0 + kh);
    acc0 = wmma_f32(a, b0, acc0);
    acc1 = wmma_f32(a, b1, acc1);
  }
  const int mbase = m0 + wave * 16 + 8 * (lane >> 4);
  float* o0 = out + n0 + nl;
  float* o1 = out + n0 + 16 + nl;
#pragma unroll
  for (int vi = 0; vi < 8; vi++) {
    o0[(mbase + vi) * Cdim] = acc0[vi];
    o1[(mbase + vi) * Cdim] = acc1[vi];
  }
}

// ---------------------------------------------------------------------------
// Chunked RWKV scan. One workgroup (512 thr / 16 waves) per batch.
// State S [256,256] f32 lives in LDS (256 KB) for the whole scan; the
// initial state image is brought in by a single TDM transfer.
// Chunk = 16 timesteps -> 4 WMMA GEMMs per chunk.
// ---------------------------------------------------------------------------
__global__ __launch_bounds__(512) void scan_kernel(const float* __restrict__ Kg,
                                                   const float* __restrict__ Vg,
                                                   const float* __restrict__ Rg,
                                                   const float* __restrict__ td,
                                                   const float* __restrict__ Sin,
                                                   float* __restrict__ Yg,
                                                   float* __restrict__ Sout) {
  extern __shared__ float smem[];
  float* S  = smem;                      // 65536 : state [c][d]
  float* Q  = S + 65536;                 // 4096  : q~ [i][c]
  float* wp = Q + 4096;                  // 17*256: w^k
  float* wr = wp + 17 * 256;             // 17*256: w^-k
  float* Am = wr + 17 * 256;             // 256   : A matrix [i][j]

  const int b = blockIdx.x;
  const int tid = threadIdx.x;
  const float* kb = Kg + b * Tdim * Cdim;
  const float* vb = Vg + b * Tdim * Cdim;
  const float* rb = Rg + b * Tdim * Cdim;

  // initial state image: one TDM DMA (256x256 f32 = 256 KB)
  if (tid < 32) {
    tdm_load_2d_f32(lds_offset_of(S), Sin + (size_t)b * 65536,
                    /*tile*/ 256, 256, /*tensor*/ 256, 256, /*stride0*/ 256);
  }
  // decay power tables (overlaps with the DMA)
  if (tid < 256) {
    float w = __expf(-__expf(td[tid]));
    float p = 1.f;
    wp[tid] = 1.f;
    wr[tid] = 1.f;
    for (int k = 1; k <= 16; k++) {
      p *= w;
      wp[k * 256 + tid] = p;
      wr[k * 256 + tid] = 1.f / p;
    }
  }
  if (tid < 32) __builtin_amdgcn_s_wait_tensorcnt(0);
  __syncthreads();

  const int wave = tid >> 5, lane = tid & 31;
  const int nl = lane & 15;
  const int kh = (lane >> 4) * 2;
  const int d0 = wave * 16;   // this wave's n-tile (d)
  const int c0 = wave * 16;   // this wave's K-slice / c-tile

  // hoist w^16 for this wave's state rows
  float w16[8];
#pragma unroll
  for (int vi = 0; vi < 8; vi++) w16[vi] = wp[16 * 256 + c0 + vi + 8 * (lane >> 4)];

  for (int ch = 0; ch < 32; ch++) {
    const int t0 = ch * 16;

    // ---- Phase B: build q~[i][c] = sigmoid(r)*w^(i+1); zero A ----
    for (int e = tid; e < 4096; e += 512) {
      int i = e >> 8, c = e & 255;
      float rv = rb[(t0 + i) * Cdim + c];
      float q = 1.f / (1.f + __expf(-rv));
      Q[e] = q * wp[(i + 1) * 256 + c];
    }
    if (tid < 256) Am[tid] = 0.f;
    __syncthreads();

    // ---- Phase C: GEMM1 (Y1 = Q~ x S), GEMM2 (A = Q~ x K^) ----
    v8f y1 = {};
#pragma unroll 4
    for (int k0 = 0; k0 < 256; k0 += 4) {
      v2f a = *(const v2f*)(Q + nl * Cdim + k0 + kh);
      v2f bm;
      bm.x = S[(k0 + kh) * Cdim + d0 + nl];
      bm.y = S[(k0 + kh + 1) * Cdim + d0 + nl];
      y1 = wmma_f32(a, bm, y1);
    }
    // GEMM2: this wave reduces K-slice [c0, c0+16)
    v8f pa = {};
#pragma unroll
    for (int k0 = 0; k0 < 16; k0 += 4) {
      int cc = c0 + k0 + kh;
      v2f a = *(const v2f*)(Q + nl * Cdim + cc);
      int j = nl;  // B column = timestep j
      v2f bm;
      bm.x = kb[(t0 + j) * Cdim + cc] * wr[(j + 1) * 256 + cc];
      bm.y = kb[(t0 + j) * Cdim + cc + 1] * wr[(j + 1) * 256 + cc + 1];
      pa = wmma_f32(a, bm, pa);
    }
    {
      int rbase = 8 * (lane >> 4);
#pragma unroll
      for (int vi = 0; vi < 8; vi++)
        atomicAdd(&Am[(rbase + vi) * 16 + nl], pa[vi]);
    }
    __syncthreads();

    // ---- Phase D: causal mask, GEMM3 (Y += Amask x V), write output ----
    if (tid < 256) {
      int i = tid >> 4, j = tid & 15;
      if (j > i) Am[tid] = 0.f;
    }
    __syncthreads();

    v8f y = y1;
#pragma unroll
    for (int k0 = 0; k0 < 16; k0 += 4) {
      int jj = k0 + kh;
      v2f a = *(const v2f*)(Am + nl * 16 + jj);
      v2f bm;
      bm.x = vb[(t0 + jj) * Cdim + d0 + nl];
      bm.y = vb[(t0 + jj + 1) * Cdim + d0 + nl];
      y = wmma_f32(a, bm, y);
    }
    {
      float* yo = Yg + (b * Tdim + t0) * Cdim + d0 + nl;
      const int ib = 8 * (lane >> 4);
#pragma unroll
      for (int vi = 0; vi < 8; vi++) yo[(ib + vi) * Cdim] = y[vi];
    }

    // ---- Phase E: S = w^16 (.) S + K^T_bar x V  (wave owns rows c0..c0+15)
    for (int dd = 0; dd < 256; dd += 16) {
      v8f acc = {};
#pragma unroll
      for (int k0 = 0; k0 < 16; k0 += 4) {
        int j0 = k0 + kh;
        int c = c0 + nl;
        v2f a;
        a.x = kb[(t0 + j0) * Cdim + c] * wp[(15 - j0) * 256 + c];
        a.y = kb[(t0 + j0 + 1) * Cdim + c] * wp[(14 - j0) * 256 + c];
        v2f bm;
        bm.x = vb[(t0 + j0) * Cdim + dd + nl];
        bm.y = vb[(t0 + j0 + 1) * Cdim + dd + nl];
        acc = wmma_f32(a, bm, acc);
      }
#pragma unroll
      for (int vi = 0; vi < 8; vi++) {
        int c = c0 + vi + 8 * (lane >> 4);
        int idx = c * Cdim + dd + nl;
        S[idx] = S[idx] * w16[vi] + acc[vi];
      }
    }
    __syncthreads();
  }

  // write final state
  {
    float4* dst = (float4*)(Sout + b * 65536);
    const float4* src = (const float4*)S;
    for (int i = tid; i < 16384; i += 512) dst[i] = src[i];
  }
}

// ---------------------------------------------------------------------------
extern "C" void kernel_launch(void* const* d_in, const int* in_sizes, int n_in,
                              void* d_out, int out_size, void* d_ws, size_t ws_size,
                              hipStream_t stream) {
  (void)in_sizes; (void)n_in; (void)out_size; (void)ws_size;
  const float* x     = (const float*)d_in[0];
  const float* state = (const float*)d_in[1];
  const float* gamma = (const float*)d_in[2];
  const float* beta  = (const float*)d_in[3];
  const float* Wk    = (const float*)d_in[4];
  const float* Wv    = (const float*)d_in[5];
  const float* Wr    = (const float*)d_in[6];
  const float* Wo    = (const float*)d_in[7];
  const float* td    = (const float*)d_in[8];
  float* out = (float*)d_out;
  float* ws  = (float*)d_ws;

  float* xn = ws;
  float* k  = ws + (size_t)BTC;
  float* v  = ws + (size_t)2 * BTC;
  float* r  = ws + (size_t)3 * BTC;
  float* y  = ws + (size_t)4 * BTC;

  ln_kernel<<<Bdim * Tdim, 256, 0, stream>>>(x, gamma, beta, xn);

  dim3 ggrid(Bdim * Tdim / 128, Cdim / 32);
  gemm_xwt<<<ggrid, 256, 0, stream>>>(xn, Wk, k);
  gemm_xwt<<<ggrid, 256, 0, stream>>>(xn, Wv, v);
  gemm_xwt<<<ggrid, 256, 0, stream>>>(xn, Wr, r);

  size_t smem = (size_t)(65536 + 4096 + 2 * 17 * 256 + 256) * sizeof(float);
  (void)hipFuncSetAttribute((const void*)scan_kernel,
                            hipFuncAttributeMaxDynamicSharedMemorySize,
                            (int)smem);
  scan_kernel<<<Bdim, 512, smem, stream>>>(k, v, r, td, state, y, out + BTC);

  gemm_xwt<<<ggrid, 256, 0, stream>>>(y, Wo, out);
}